// STAttentionBlock_75969381531868
// MI455X (gfx1250) — compile-verified
//
#include <hip/hip_runtime.h>
#include <hip/hip_bf16.h>
#include <math.h>

#define NB   32
#define CCH  64
#define TT   400
#define VVN  27
#define PPN  (TT*VVN)   // 10800
#define SSH  2
#define ICH  16
#define EPSV 1e-5f

typedef float v2f __attribute__((ext_vector_type(2)));
typedef float v8f __attribute__((ext_vector_type(8)));

// D = A(16x4 f32) * B(4x16 f32) + C(16x16 f32)
static __device__ __forceinline__ v8f wmma4(v2f a, v2f b, v8f c) {
  return __builtin_amdgcn_wmma_f32_16x16x4_f32(false, a, false, b, (short)0, c,
                                               false, false);
}

// ---------------- spatial positional encoding (64 x 27) ----------------
__global__ void k_pe(float* __restrict__ pe) {
  int i = blockIdx.x * blockDim.x + threadIdx.x;
  if (i < CCH * VVN) {
    int c = i / VVN, v = i % VVN;
    float div = expf(-logf(10000.f) * (float)(2 * (c / 2)) / (float)CCH);
    float ang = div * (float)v;
    pe[i] = (c & 1) ? cosf(ang) : sinf(ang);
  }
}

// ---------------- generic 64-out-channel conv1x1 GEMM -------------------
// out[n,o,p] = epilogue( sum_c W[o, wcol0+c] * (in[n,c,p] (+ pe[c, p%27])) )
// Tile: 64 M x 32 P per block, 8 waves (4 Mtiles x 2 Ptiles), K=64 via WMMA.
__global__ void __launch_bounds__(256)
k_gemm64(const float* __restrict__ in, const float* __restrict__ W,
         int ldw, int wcol0,
         const float* __restrict__ bias, const float* __restrict__ pe,
         const float* __restrict__ addprev,
         const float* __restrict__ bng, const float* __restrict__ bnb,
         const float* __restrict__ bnm, const float* __restrict__ bnv,
         const float* __restrict__ resid, int do_leaky,
         float* __restrict__ out) {
  __shared__ float wlds[64 * 65];   // padded stride vs 64 banks
  __shared__ float xlds[64 * 33];
  const int n = blockIdx.y;
  const int pbase = blockIdx.x * 32;
  const int tid = threadIdx.x;

  for (int i = tid; i < 64 * 64; i += 256) {
    int o = i >> 6, c = i & 63;
    wlds[o * 65 + c] = W[o * ldw + wcol0 + c];
  }
  const float* inb = in + (size_t)n * 64 * PPN;
  for (int i = tid; i < 64 * 32; i += 256) {
    int c = i >> 5, pp = i & 31;
    int p = pbase + pp;
    float x = 0.f;
    if (p < PPN) {
      x = inb[(size_t)c * PPN + p];
      if (pe) x += pe[c * VVN + (p % VVN)];
    }
    xlds[c * 33 + pp] = x;
  }
  __syncthreads();

  const int wave = tid >> 5, lane = tid & 31;
  const int half = lane >> 4, lm = lane & 15;
  const int m0 = (wave & 3) * 16;
  const int pt = (wave >> 2) * 16;
  v8f acc = {0.f, 0.f, 0.f, 0.f, 0.f, 0.f, 0.f, 0.f};
#pragma unroll
  for (int kk = 0; kk < 64; kk += 4) {
    int k = kk + 2 * half;
    v2f a, b;
    a.x = wlds[(m0 + lm) * 65 + k];
    a.y = wlds[(m0 + lm) * 65 + k + 1];
    b.x = xlds[k * 33 + pt + lm];
    b.y = xlds[(k + 1) * 33 + pt + lm];
    acc = wmma4(a, b, acc);
  }

  int p = pbase + pt + lm;
  if (p < PPN) {
    size_t base = (size_t)n * 64 * PPN;
#pragma unroll
    for (int r = 0; r < 8; ++r) {
      int o = m0 + r + 8 * half;
      float v = acc[r];
      size_t idx = base + (size_t)o * PPN + p;
      if (addprev) v += addprev[idx];
      if (bias) v += bias[o];
      if (bng) {
        float inv = bng[o] * rsqrtf(bnv[o] + EPSV);
        v = v * inv + (bnb[o] - bnm[o] * inv);
      }
      if (resid) v += resid[idx];
      if (do_leaky) v = (v >= 0.f) ? v : 0.1f * v;
      out[idx] = v;
    }
  }
}

// ---------------- attention-score partials ------------------------------
// partial[ns,tc,u,v] = sum over c(16), t in chunk(50) of q[c,t,u]*k[c,t,v]
// K ordered as k = t*16 + c, chunks of 4 never cross t. 27x27 padded to 32x32;
// garbage rows/cols are discarded at the reduction step.
__global__ void __launch_bounds__(128)
k_att_partial(const float* __restrict__ qk, float* __restrict__ partial) {
  const int ns = blockIdx.x;           // 0..63 (n*2+s)
  const int tc = blockIdx.y;           // 0..7
  const int n = ns >> 1, s = ns & 1;
  const float* qb = qk + (size_t)n * 64 * PPN + (size_t)(s * ICH) * PPN;
  const float* kb = qb + (size_t)(SSH * ICH) * PPN;   // +32 channels
  const int tid = threadIdx.x;
  const int wave = tid >> 5, lane = tid & 31;
  const int half = lane >> 4, lm = lane & 15;
  const int u0 = (wave & 1) * 16, v0 = (wave >> 1) * 16;
  v8f acc = {0.f, 0.f, 0.f, 0.f, 0.f, 0.f, 0.f, 0.f};
  const int kbeg = tc * 800;           // 50 t-steps * 16 channels
  for (int kk = 0; kk < 800; kk += 4) {
    int k = kbeg + kk + 2 * half;      // c = k&15 (even), t = k>>4
    int c = k & 15, t = k >> 4;
    size_t off = (size_t)c * PPN + t * VVN;
    v2f a, b;
    a.x = qb[off + u0 + lm];
    a.y = qb[off + PPN + u0 + lm];     // channel c+1, same t
    b.x = kb[off + v0 + lm];
    b.y = kb[off + PPN + v0 + lm];
    acc = wmma4(a, b, acc);
  }
  float* pt_ = partial + ((size_t)(ns * 8 + tc)) * 1024;
#pragma unroll
  for (int r = 0; r < 8; ++r) {
    int u = u0 + r + 8 * half, v = v0 + lm;
    pt_[u * 32 + v] = acc[r];
  }
}

__global__ void k_att_final(const float* __restrict__ partial,
                            const float* __restrict__ alphas,
                            const float* __restrict__ att0,
                            float* __restrict__ att) {
  int i = blockIdx.x * blockDim.x + threadIdx.x;
  if (i >= 64 * VVN * VVN) return;
  int ns = i / (VVN * VVN);
  int uv = i % (VVN * VVN);
  int u = uv / VVN, v = uv % VVN;
  int s = ns & 1;
  float sum = 0.f;
  for (int tc = 0; tc < 8; ++tc)
    sum += partial[((size_t)(ns * 8 + tc)) * 1024 + u * 32 + v];
  att[(size_t)ns * VVN * VVN + uv] =
      tanhf(sum / (float)(ICH * TT)) * alphas[s] + att0[(s * VVN + u) * VVN + v];
}

// ---------------- aggregation: agg[n,c,t,v] = sum_u x[n,c,t,u]*att[n,s,u,v]
// Rows (c*T+t) are contiguous blocks of 27 floats; K=27 padded to 28 with a
// zeroed B row so the out-of-row A element contributes nothing.
__global__ void __launch_bounds__(256)
k_agg(const float* __restrict__ x, const float* __restrict__ att, int s,
      float* __restrict__ out) {
  __shared__ float xl[64 * 28];
  __shared__ float bl[28 * 28];
  const int n = blockIdx.y;
  const int row0 = blockIdx.x * 64;    // of 25600 rows per n
  const int tid = threadIdx.x;
  const float* xb = x + (size_t)n * 64 * PPN;
  for (int i = tid; i < 64 * 27; i += 256) {
    int r = i / 27, u = i % 27;
    xl[r * 28 + u] = xb[(size_t)row0 * 27 + i];
  }
  if (tid < 64) xl[tid * 28 + 27] = 0.f;
  const float* ab = att + ((size_t)(n * 2 + s)) * VVN * VVN;
  for (int i = tid; i < 27 * 27; i += 256)
    bl[(i / 27) * 28 + (i % 27)] = ab[i];
  if (tid < 28) { bl[27 * 28 + tid] = 0.f; bl[tid * 28 + 27] = 0.f; }
  __syncthreads();

  const int wave = tid >> 5, lane = tid & 31;
  const int half = lane >> 4, lm = lane & 15;
  const int mt = (wave & 3) * 16;
  const int v0 = (wave >> 2) * 16;
  v8f acc = {0.f, 0.f, 0.f, 0.f, 0.f, 0.f, 0.f, 0.f};
#pragma unroll
  for (int kk = 0; kk < 28; kk += 4) {
    int k = kk + 2 * half;
    v2f a, b;
    a.x = xl[(mt + lm) * 28 + k];
    a.y = xl[(mt + lm) * 28 + k + 1];
    b.x = bl[k * 28 + v0 + lm];
    b.y = bl[(k + 1) * 28 + v0 + lm];
    acc = wmma4(a, b, acc);
  }
  int v = v0 + lm;
  if (v < VVN) {
#pragma unroll
    for (int r = 0; r < 8; ++r) {
      int row = row0 + mt + r + 8 * half;
      out[(size_t)n * 64 * PPN + (size_t)row * 27 + v] = acc[r];
    }
  }
}

// ---------------- temporal 3x1 conv as K=192 im2col GEMM ----------------
__global__ void __launch_bounds__(256)
k_tconv(const float* __restrict__ y2, const float* __restrict__ wt,
        const float* __restrict__ bt, const float* __restrict__ g,
        const float* __restrict__ bb, const float* __restrict__ mm,
        const float* __restrict__ vv, float* __restrict__ out) {
  __shared__ float xl[64 * 87];        // window [pbase-27, pbase+59)
  const int n = blockIdx.y;
  const int pbase = blockIdx.x * 32;
  const int tid = threadIdx.x;
  const float* yb = y2 + (size_t)n * 64 * PPN;
  for (int i = tid; i < 64 * 86; i += 256) {
    int c = i / 86, j = i % 86;
    int p = pbase - 27 + j;
    xl[c * 87 + j] = (p >= 0 && p < PPN) ? yb[(size_t)c * PPN + p] : 0.f;
  }
  __syncthreads();

  const int wave = tid >> 5, lane = tid & 31;
  const int half = lane >> 4, lm = lane & 15;
  const int m0 = (wave & 3) * 16;
  const int pt = (wave >> 2) * 16;
  v8f acc = {0.f, 0.f, 0.f, 0.f, 0.f, 0.f, 0.f, 0.f};
  for (int kk = 0; kk < 192; kk += 4) {
    int k = kk + 2 * half;             // k = c*3 + kt
    v2f a, b;
    a.x = wt[(m0 + lm) * 192 + k];     // weights stay in L2/L0 (48 KB)
    a.y = wt[(m0 + lm) * 192 + k + 1];
    b.x = xl[(k / 3) * 87 + (k % 3) * 27 + pt + lm];
    int k1 = k + 1;
    b.y = xl[(k1 / 3) * 87 + (k1 % 3) * 27 + pt + lm];
    acc = wmma4(a, b, acc);
  }
  int p = pbase + pt + lm;
  if (p < PPN) {
    size_t base = (size_t)n * 64 * PPN;
#pragma unroll
    for (int r = 0; r < 8; ++r) {
      int o = m0 + r + 8 * half;
      float inv = g[o] * rsqrtf(vv[o] + EPSV);
      float val = acc[r] + bt[o];
      val = val * inv + (bb[o] - mm[o] * inv);
      size_t idx = base + (size_t)o * PPN + p;
      val += y2[idx];
      out[idx] = (val >= 0.f) ? val : 0.1f * val;
    }
  }
}

extern "C" void kernel_launch(void* const* d_in, const int* in_sizes, int n_in,
                              void* d_out, int out_size, void* d_ws,
                              size_t ws_size, hipStream_t stream) {
  const float* x      = (const float*)d_in[0];
  const float* w_in   = (const float*)d_in[1];
  const float* b_in   = (const float*)d_in[2];
  const float* alphas = (const float*)d_in[3];
  const float* att0   = (const float*)d_in[4];
  const float* w_outs = (const float*)d_in[5];
  const float* b_outs = (const float*)d_in[6];
  const float* bog = (const float*)d_in[7],  *bob = (const float*)d_in[8];
  const float* bom = (const float*)d_in[9],  *bov = (const float*)d_in[10];
  const float* w_ff = (const float*)d_in[11], *b_ff = (const float*)d_in[12];
  const float* bfg = (const float*)d_in[13], *bfb = (const float*)d_in[14];
  const float* bfm = (const float*)d_in[15], *bfv = (const float*)d_in[16];
  const float* w_t = (const float*)d_in[17], *b_t = (const float*)d_in[18];
  const float* btg = (const float*)d_in[19], *btb = (const float*)d_in[20];
  const float* btm = (const float*)d_in[21], *btv = (const float*)d_in[22];
  float* out = (float*)d_out;

  float* ws = (float*)d_ws;
  const size_t SZ = (size_t)NB * 64 * PPN;       // 22,118,400 floats
  float* buf0    = ws;                            // qk -> agg_s -> y2
  float* att     = ws + SZ + 1024;                // pad for att-partial overreads
  float* partial = att + 64 * VVN * VVN;          // 64*8 tiles of 32x32
  float* pe      = partial + (size_t)64 * 8 * 1024;

  dim3 gg((PPN + 31) / 32, NB);

  k_pe<<<dim3((CCH * VVN + 255) / 256), 256, 0, stream>>>(pe);

  // qk = W_in @ (x + pe) + b_in  -> buf0
  k_gemm64<<<gg, 256, 0, stream>>>(x, w_in, 64, 0, b_in, pe, nullptr, nullptr,
                                   nullptr, nullptr, nullptr, nullptr, 0, buf0);
  // attention scores
  k_att_partial<<<dim3(64, 8), 128, 0, stream>>>(buf0, partial);
  k_att_final<<<(64 * VVN * VVN + 255) / 256, 256, 0, stream>>>(partial, alphas,
                                                                att0, att);
  // head 0: out = W0 @ (x . att0)
  k_agg<<<dim3(400, NB), 256, 0, stream>>>(x, att, 0, buf0);
  k_gemm64<<<gg, 256, 0, stream>>>(buf0, w_outs, 128, 0, nullptr, nullptr,
                                   nullptr, nullptr, nullptr, nullptr, nullptr,
                                   nullptr, 0, out);
  // head 1 + bias + BN + residual(x) + leaky  -> out (= y)
  k_agg<<<dim3(400, NB), 256, 0, stream>>>(x, att, 1, buf0);
  k_gemm64<<<gg, 256, 0, stream>>>(buf0, w_outs, 128, 64, b_outs, nullptr, out,
                                   bog, bob, bom, bov, x, 1, out);
  // feed-forward: buf0 (= y2) = leaky(x + BN(W_ff @ y + b_ff))
  k_gemm64<<<gg, 256, 0, stream>>>(out, w_ff, 64, 0, b_ff, nullptr, nullptr,
                                   bfg, bfb, bfm, bfv, x, 1, buf0);
  // temporal conv + BN + residual(y2) + leaky -> out (= z)
  k_tconv<<<gg, 256, 0, stream>>>(buf0, w_t, b_t, btg, btb, btm, btv, out);
}